// Dense3DSpatialTransformer_11630771437796
// MI455X (gfx1250) — compile-verified
//
#include <hip/hip_runtime.h>
#include <stdint.h>

// Dense 2D spatial transformer (bilinear warp) for MI455X / gfx1250.
// Bandwidth-bound: ~268 MB min HBM traffic -> ~11.5 us floor @ 23.3 TB/s.
//  * flow field staged via gfx1250 async VMEM->LDS DMA (ASYNCcnt path),
//    read back per-lane with ds_load_b128 (no cross-lane sharing, no barrier)
//  * image gathers regular-temporal (64 MiB image stays resident in 192 MB L2)
//  * output streamed with non-temporal b128 stores
//  * global_prefetch_b8 ahead on the flow streams
//  * wave32-friendly 256-thread blocks, 4 pixels/thread

typedef float v4f __attribute__((ext_vector_type(4)));

#define HH 4096
#define WW 4096

// Fetch from the conceptually zero-padded image: padded coords (r,c) in
// [0, HH+1] x [0, WW+1]; interior (r,c) maps to img[r-1][c-1], border -> 0.
__device__ __forceinline__ float img_fetch(const float* __restrict__ img,
                                           int r, int c) {
  const bool inside = ((unsigned)(r - 1) < (unsigned)HH) &
                      ((unsigned)(c - 1) < (unsigned)WW);
  float v = 0.0f;
  if (inside) {
    v = img[(size_t)(r - 1) * WW + (unsigned)(c - 1)];
  }
  return v;
}

__device__ __forceinline__ float bilerp(const float* __restrict__ img,
                                        float Hu, float Wu) {
  // floor BEFORE clipping (matches reference); int conversion is exact
  // because the value is already floored.
  int hf = (int)floorf(Hu);
  int wf = (int)floorf(Wu);
  int hc = hf + 1;
  int wc = wf + 1;
  hf = min(max(hf, 0), HH + 1);
  hc = min(max(hc, 0), HH + 1);
  wf = min(max(wf, 0), WW + 1);
  wc = min(max(wc, 0), WW + 1);

  // weights from CLIPPED ceil coords (matches reference exactly)
  const float dH = (float)hc - Hu;
  const float dW = (float)wc - Wu;
  const float oH = 1.0f - dH;
  const float oW = 1.0f - dW;

  const float v00 = img_fetch(img, hf, wf);
  const float v10 = img_fetch(img, hc, wf);
  const float v01 = img_fetch(img, hf, wc);
  const float v11 = img_fetch(img, hc, wc);

  return v00 * (dW * dH) + v10 * (dW * oH) + v01 * (oW * dH) + v11 * (oW * oH);
}

// Low 32 bits of a generic pointer into LDS aperture == DS-space byte address
// (the flat LDS aperture maps by the low 32 address bits).
__device__ __forceinline__ uint32_t lds_byte_addr(const void* p) {
  return (uint32_t)(uintptr_t)p;
}

__global__ __launch_bounds__(256) void warp_bilinear_kernel(
    const float* __restrict__ img,   // [H, W]
    const float* __restrict__ flow,  // [2, H, W] (ch0 = dH, ch1 = dW)
    float* __restrict__ out)         // [H, W]
{
  constexpr int W4 = WW / 4;  // 4 pixels per thread along W
  const unsigned tid = blockIdx.x * blockDim.x + threadIdx.x;
  const int h  = (int)(tid / W4);
  const int w0 = (int)(tid % W4) * 4;
  const size_t base = (size_t)h * WW + (size_t)w0;

  const float* __restrict__ flowH = flow;
  const float* __restrict__ flowW = flow + (size_t)HH * WW;

  // Per-lane staging slots for the two flow channels (16 B each).
  __shared__ v4f s_fh[256];
  __shared__ v4f s_fw[256];

  // gfx1250 async VMEM->LDS DMA: each lane pulls its own float4 of each flow
  // channel straight into LDS (no VGPR writeback). Tracked by ASYNCcnt.
  {
    const uint32_t lh = lds_byte_addr(&s_fh[threadIdx.x]);
    const uint32_t lw = lds_byte_addr(&s_fw[threadIdx.x]);
    const float* gh = flowH + base;
    const float* gw = flowW + base;
    asm volatile(
        "global_load_async_to_lds_b128 %0, %2, off\n\t"
        "global_load_async_to_lds_b128 %1, %3, off"
        :
        : "v"(lh), "v"(lw), "v"(gh), "v"(gw)
        : "memory");
  }

  // gfx1250 prefetch (global_prefetch_b8), ~8KB ahead on each flow stream;
  // speculative OOB prefetch at the tail is silently dropped.
  if ((threadIdx.x & 63u) == 0u) {
    __builtin_prefetch(flowH + base + 2048, 0, 1);
    __builtin_prefetch(flowW + base + 2048, 0, 1);
  }

  // Overlap: do the coordinate bases while the DMA is in flight.
  const float hb = (float)h + 1.0f;   // +1 for the zero-pad border shift
  const float wb = (float)w0 + 1.0f;

  // Drain the async engine, then read our own slots back (same-lane data:
  // no cross-wave sharing, so no workgroup barrier is required).
  asm volatile("s_wait_asynccnt 0x0" ::: "memory");
  const v4f fh = s_fh[threadIdx.x];
  const v4f fw = s_fw[threadIdx.x];

  v4f o;
  o.x = bilerp(img, fh.x + hb, fw.x + wb + 0.0f);
  o.y = bilerp(img, fh.y + hb, fw.y + wb + 1.0f);
  o.z = bilerp(img, fh.z + hb, fw.z + wb + 2.0f);
  o.w = bilerp(img, fh.w + hb, fw.w + wb + 3.0f);

  // Non-temporal b128 store: output is written once, never re-read.
  __builtin_nontemporal_store(o, (v4f*)(out + base));
}

extern "C" void kernel_launch(void* const* d_in, const int* in_sizes, int n_in,
                              void* d_out, int out_size, void* d_ws, size_t ws_size,
                              hipStream_t stream) {
  (void)in_sizes; (void)n_in; (void)out_size; (void)d_ws; (void)ws_size;

  const float* img  = (const float*)d_in[0];  // input1: [1,1,4096,4096] fp32
  const float* flow = (const float*)d_in[1];  // input2: [1,2,4096,4096] fp32
  float* out        = (float*)d_out;          // [1,1,4096,4096] fp32

  constexpr int threads = 256;                       // 8 wave32s per block
  constexpr int total   = (HH * WW) / 4;             // 4 px per thread
  constexpr int blocks  = total / threads;           // 16384 blocks

  warp_bilinear_kernel<<<blocks, threads, 0, stream>>>(img, flow, out);
}